// CrossAttentionModule_10840497455895
// MI455X (gfx1250) — compile-verified
//
#include <hip/hip_runtime.h>
#include <hip/hip_bf16.h>

// ---------------------------------------------------------------------------
// CLS-token cross attention, refactored so K/V are never materialized:
//   t[b,h,c]  = SCALE * sum_d W_kv[c, h*64+d] * q[b,h,d]
//   s[b,h,n]  = sum_c X[b,n,c] * t[b,h,c]            (WMMA f32 16x16x4)
//   attn      = softmax_n(s)
//   u[b,h,c]  = sum_n attn[b,h,n] * X[b,n,c]         (WMMA f32 16x16x4)
//   out[b,hd] = sum_c u[b,h,c] * W_kv[c, 768+hd]
//   y         = (x_cls_p + out) @ W_back + b_back
// ---------------------------------------------------------------------------

typedef __attribute__((ext_vector_type(2))) float v2f;
typedef __attribute__((ext_vector_type(8))) float v8f;

#define B_SZ   64
#define NIMG   576
#define NTOK   577          // 1 cls + 576 img tokens
#define NPAD   580          // attn row stride (NTOK padded to multiple of 4)
#define CCLS   384
#define CDIM   768
#define C2     1536
#define HN     12
#define DH     64
#define SCL    0.125f       // 1/sqrt(64)
#define NTILES 37           // ceil(577/16)
#define CTILES 48           // 768/16

// ------------------------- tiny VALU glue kernels --------------------------

__global__ void k_zero(float* __restrict__ p, int n) {
  for (int i = threadIdx.x; i < n; i += 256) p[i] = 0.f;
}

// x_cls_p[b,c] = x_cls[b,:] @ W_proj[:,c] + b_proj[c]
__global__ void k_proj_cls(const float* __restrict__ x_cls,
                           const float* __restrict__ W_proj,
                           const float* __restrict__ b_proj,
                           float* __restrict__ xclsp) {
  int idx = blockIdx.x * blockDim.x + threadIdx.x;      // b*768 + c
  int b = idx / CDIM, c = idx % CDIM;
  const float* xr = x_cls + (size_t)b * CCLS;
  float acc = b_proj[c];
  #pragma unroll 4
  for (int k = 0; k < CCLS; ++k) acc = fmaf(xr[k], W_proj[(size_t)k * CDIM + c], acc);
  xclsp[idx] = acc;
}

// q[b,c] = x_cls_p[b,:] @ W_q[:,c]
__global__ void k_q(const float* __restrict__ xclsp,
                    const float* __restrict__ W_q,
                    float* __restrict__ q) {
  int idx = blockIdx.x * blockDim.x + threadIdx.x;      // b*768 + c
  int b = idx / CDIM, c = idx % CDIM;
  const float* xr = xclsp + (size_t)b * CDIM;
  float acc = 0.f;
  #pragma unroll 4
  for (int k = 0; k < CDIM; ++k) acc = fmaf(xr[k], W_q[(size_t)k * CDIM + c], acc);
  q[idx] = acc;
}

// t[b,h,c] = SCALE * sum_d q[b,h*64+d] * W_kv[c, h*64+d]
__global__ void k_t(const float* __restrict__ q,
                    const float* __restrict__ W_kv,
                    float* __restrict__ t) {
  int idx = blockIdx.x * blockDim.x + threadIdx.x;      // (b*12+h)*768 + c
  int c = idx % CDIM;
  int bh = idx / CDIM;
  int h = bh % HN, b = bh / HN;
  const float* qr = q + (size_t)b * CDIM + h * DH;
  const float* wr = W_kv + (size_t)c * C2 + h * DH;     // K half of W_kv
  float acc = 0.f;
  #pragma unroll 8
  for (int d = 0; d < DH; ++d) acc = fmaf(qr[d], wr[d], acc);
  t[idx] = acc * SCL;
}

// -------------------------- WMMA f32 bulk kernels ---------------------------
// Fragment layout per CDNA5 ISA (wave32):
//   A 16x4 : lane&15 = M row; lanes 0-15 hold K={0,1}, lanes 16-31 K={2,3}
//   B 4x16 : lane&15 = N col; lanes 0-15 hold K={0,1}, lanes 16-31 K={2,3}
//   C 16x16: VGPR j holds C[j + 8*(lane>>4)][lane&15]

// scores: per (b): S(577x12) = X(577x768) @ T(768x12).  M=tok, N=h, K=c.
__global__ void k_scores(const float* __restrict__ xclsp,
                         const float* __restrict__ ximg,
                         const float* __restrict__ t,
                         const float* __restrict__ zpage,
                         float* __restrict__ attn) {
  const int tile = blockIdx.x;          // token tile 0..36
  const int b    = blockIdx.y;
  const int lane = threadIdx.x & 31;
  const int half = lane >> 4;
  const int lm   = lane & 15;

  const int tok = tile * 16 + lm;       // A row (M)

  // loop-invariant per-lane row pointers; out-of-range lanes -> zero page
  const float* arow = zpage;
  if (tok == 0)         arow = xclsp + (size_t)b * CDIM;
  else if (tok < NTOK)  arow = ximg + ((size_t)b * NIMG + (tok - 1)) * CDIM;
  const float* brow = (lm < HN) ? (t + ((size_t)b * HN + lm) * CDIM) : zpage;
  arow += 2 * half;                     // fold K sub-pair offset into base
  brow += 2 * half;

  v8f acc = {0.f, 0.f, 0.f, 0.f, 0.f, 0.f, 0.f, 0.f};
  #pragma unroll 4
  for (int k0 = 0; k0 < CDIM; k0 += 4) {
    v2f a  = *(const v2f*)(arow + k0);
    v2f bb = *(const v2f*)(brow + k0);
    acc = __builtin_amdgcn_wmma_f32_16x16x4_f32(false, a, false, bb,
                                                (short)0, acc, false, false);
  }
  #pragma unroll
  for (int j = 0; j < 8; ++j) {
    const int m    = j + 8 * half;
    const int tok2 = tile * 16 + m;
    if (tok2 < NTOK && lm < HN)
      attn[((size_t)b * HN + lm) * NPAD + tok2] = acc[j];
  }
}

// row-wise softmax over n=577 for each (b,h); zero the 3 pad slots
__global__ void k_softmax(float* __restrict__ attn) {
  float* row = attn + (size_t)blockIdx.x * NPAD;
  __shared__ float sred[4];
  const int tid = threadIdx.x;                 // blockDim = 128 (4 waves)
  float m = -3.0e38f;
  for (int i = tid; i < NTOK; i += 128) m = fmaxf(m, row[i]);
  for (int o = 16; o > 0; o >>= 1) m = fmaxf(m, __shfl_xor(m, o, 32));
  if ((tid & 31) == 0) sred[tid >> 5] = m;
  __syncthreads();
  m = fmaxf(fmaxf(sred[0], sred[1]), fmaxf(sred[2], sred[3]));
  __syncthreads();
  float s = 0.f;
  for (int i = tid; i < NTOK; i += 128) {
    float e = expf(row[i] - m);
    row[i] = e;
    s += e;
  }
  for (int o = 16; o > 0; o >>= 1) s += __shfl_xor(s, o, 32);
  if ((tid & 31) == 0) sred[tid >> 5] = s;
  __syncthreads();
  const float inv = 1.0f / (sred[0] + sred[1] + sred[2] + sred[3]);
  for (int i = tid; i < NTOK; i += 128) row[i] *= inv;
  if (tid < NPAD - NTOK) row[NTOK + tid] = 0.f;  // zero K padding
}

// u: per (b): U(12x768) = attn(12x577) @ X(577x768).  M=h, N=c, K=tok.
__global__ void k_u(const float* __restrict__ xclsp,
                    const float* __restrict__ ximg,
                    const float* __restrict__ attn,
                    const float* __restrict__ zpage,
                    float* __restrict__ u) {
  const int ct   = blockIdx.x;          // c tile 0..47
  const int b    = blockIdx.y;
  const int lane = threadIdx.x & 31;
  const int half = lane >> 4;
  const int lm   = lane & 15;

  const int c = ct * 16 + lm;           // B col (N)
  const float* arow =
      ((lm < HN) ? attn + ((size_t)b * HN + lm) * NPAD : zpage) + 2 * half;
  // xb[tok * CDIM] == X[b][tok][c] for tok >= 1 (image rows)
  const float* xb = ximg + ((size_t)b * NIMG - 1) * CDIM + c;

  v8f acc = {0.f, 0.f, 0.f, 0.f, 0.f, 0.f, 0.f, 0.f};

  // k0 = 0: tok = {0,1} (half 0) / {2,3} (half 1); tok 0 is the cls row
  {
    v2f a = *(const v2f*)(arow);
    v2f bb;
    const size_t t0 = 2 * half;
    bb.x = (half == 0) ? xclsp[(size_t)b * CDIM + c] : xb[t0 * CDIM];
    bb.y = xb[(t0 + 1) * CDIM];
    acc = __builtin_amdgcn_wmma_f32_16x16x4_f32(false, a, false, bb,
                                                (short)0, acc, false, false);
  }
  // main loop: tok = kA, kA+1 with 4 <= kA, kA+1 <= 575  (all image rows)
  #pragma unroll 4
  for (int k0 = 4; k0 < 576; k0 += 4) {
    const int kA = k0 + 2 * half;
    v2f a = *(const v2f*)(arow + k0);
    v2f bb;
    bb.x = xb[(size_t)kA * CDIM];
    bb.y = xb[(size_t)(kA + 1) * CDIM];
    acc = __builtin_amdgcn_wmma_f32_16x16x4_f32(false, a, false, bb,
                                                (short)0, acc, false, false);
  }
  // tail k0 = 576: tok up to 579; attn pad is zero there, clamp B reads
  {
    const int kA = 576 + 2 * half;
    v2f a = *(const v2f*)(arow + 576);
    v2f bb;
    const size_t t0 = (kA     <= 576) ? (size_t)kA       : 576;
    const size_t t1 = (kA + 1 <= 576) ? (size_t)(kA + 1) : 576;
    bb.x = xb[t0 * CDIM];
    bb.y = xb[t1 * CDIM];
    acc = __builtin_amdgcn_wmma_f32_16x16x4_f32(false, a, false, bb,
                                                (short)0, acc, false, false);
  }

  #pragma unroll
  for (int j = 0; j < 8; ++j) {
    const int m = j + 8 * half;          // head index
    if (m < HN)
      u[((size_t)b * HN + m) * CDIM + ct * 16 + lm] = acc[j];
  }
}

// ------------------------- output projection kernels ------------------------

// hbuf[b,cp] = x_cls_p[b,cp] + sum_c u[b, cp/64, c] * W_kv[c, 768+cp]
__global__ void k_attnout(const float* __restrict__ u,
                          const float* __restrict__ W_kv,
                          const float* __restrict__ xclsp,
                          float* __restrict__ hbuf) {
  int idx = blockIdx.x * blockDim.x + threadIdx.x;      // b*768 + cp
  int b = idx / CDIM, cp = idx % CDIM;
  int h = cp / DH;
  const float* ur = u + ((size_t)b * HN + h) * CDIM;
  float acc = 0.f;
  #pragma unroll 4
  for (int c = 0; c < CDIM; ++c)
    acc = fmaf(ur[c], W_kv[(size_t)c * C2 + CDIM + cp], acc);
  hbuf[idx] = acc + xclsp[idx];
}

// y[b,j] = hbuf[b,:] @ W_back[:,j] + b_back[j]
__global__ void k_back(const float* __restrict__ hbuf,
                       const float* __restrict__ W_back,
                       const float* __restrict__ b_back,
                       float* __restrict__ out) {
  int idx = blockIdx.x * blockDim.x + threadIdx.x;      // b*384 + j
  int b = idx / CCLS, j = idx % CCLS;
  const float* hr = hbuf + (size_t)b * CDIM;
  float acc = b_back[j];
  #pragma unroll 4
  for (int c = 0; c < CDIM; ++c) acc = fmaf(hr[c], W_back[(size_t)c * CCLS + j], acc);
  out[idx] = acc;
}

// ---------------------------------------------------------------------------

extern "C" void kernel_launch(void* const* d_in, const int* in_sizes, int n_in,
                              void* d_out, int out_size, void* d_ws, size_t ws_size,
                              hipStream_t stream) {
  const float* x_cls  = (const float*)d_in[0];
  const float* x_img  = (const float*)d_in[1];
  const float* W_proj = (const float*)d_in[2];
  const float* b_proj = (const float*)d_in[3];
  const float* W_q    = (const float*)d_in[4];
  const float* W_kv   = (const float*)d_in[5];
  const float* W_back = (const float*)d_in[6];
  const float* b_back = (const float*)d_in[7];
  float* out = (float*)d_out;

  // workspace layout (floats)
  float* ws    = (float*)d_ws;
  float* xclsp = ws;                                   // 64*768      = 49152
  float* q     = xclsp + (size_t)B_SZ * CDIM;          // 64*768      = 49152
  float* t     = q     + (size_t)B_SZ * CDIM;          // 64*12*768   = 589824
  float* attn  = t     + (size_t)B_SZ * HN * CDIM;     // 64*12*580   = 445440
  float* u     = attn  + (size_t)B_SZ * HN * NPAD;     // 64*12*768   = 589824
  float* hbuf  = u     + (size_t)B_SZ * HN * CDIM;     // 64*768      = 49152
  float* zpage = hbuf  + (size_t)B_SZ * CDIM;          // 1024 zeros
  // total ~= 7.1 MB

  k_zero<<<1, 256, 0, stream>>>(zpage, 1024);
  k_proj_cls<<<(B_SZ * CDIM) / 256, 256, 0, stream>>>(x_cls, W_proj, b_proj, xclsp);
  k_q<<<(B_SZ * CDIM) / 256, 256, 0, stream>>>(xclsp, W_q, q);
  k_t<<<(B_SZ * HN * CDIM) / 256, 256, 0, stream>>>(q, W_kv, t);
  k_scores<<<dim3(NTILES, B_SZ), 32, 0, stream>>>(xclsp, x_img, t, zpage, attn);
  k_softmax<<<B_SZ * HN, 128, 0, stream>>>(attn);
  k_u<<<dim3(CTILES, B_SZ), 32, 0, stream>>>(xclsp, x_img, attn, zpage, u);
  k_attnout<<<(B_SZ * CDIM) / 256, 256, 0, stream>>>(u, W_kv, xclsp, hbuf);
  k_back<<<(B_SZ * CCLS) / 256, 256, 0, stream>>>(hbuf, W_back, b_back, out);
}